// MyModel_3848290697240
// MI455X (gfx1250) — compile-verified
//
#include <hip/hip_runtime.h>
#include <hip/hip_bf16.h>

typedef __attribute__((ext_vector_type(2))) float v2f;
typedef __attribute__((ext_vector_type(8))) float v8f;

#define BB 384
#define DD 1024
#define WD 300
#define NN 8000
#define LL 16

static __device__ __forceinline__ v8f wmma4(v2f a, v2f b, v8f c) {
  // V_WMMA_F32_16X16X4_F32 : D = A(16x4,f32) x B(4x16,f32) + C(16x16,f32)
  return __builtin_amdgcn_wmma_f32_16x16x4_f32(false, a, false, b, (short)0, c,
                                               false, false);
}

// ---------------------------------------------------------------------------
// cls_weights[N,D] = embed_table[all_nouns_token] @ noun_W + noun_b
// One wave computes a 64x16 tile (4 M-subtiles reuse each B fragment).
// grid (N/64, D/16), block 32.
// ---------------------------------------------------------------------------
__global__ __launch_bounds__(32) void k_cls(const int* __restrict__ tok,
                                            const float* __restrict__ emb,
                                            const float* __restrict__ W,
                                            const float* __restrict__ bias,
                                            float* __restrict__ out) {
  const int lane = threadIdx.x & 31;
  const int half = lane >> 4;   // 0: K pair {0,1}, 1: K pair {2,3}
  const int lid  = lane & 15;
  const int m0 = blockIdx.x * 64, n0 = blockIdx.y * 16;
  const int col = n0 + lid;
  const float* arow[4];
#pragma unroll
  for (int s = 0; s < 4; ++s)
    arow[s] = emb + (size_t)tok[m0 + s * 16 + lid] * WD;   // gathered A rows
  v8f acc[4];
#pragma unroll
  for (int s = 0; s < 4; ++s) acc[s] = (v8f){0.f,0.f,0.f,0.f,0.f,0.f,0.f,0.f};
  for (int k0 = 0; k0 < WD; k0 += 4) {
    const int ka = k0 + 2 * half;                 // even -> 8B aligned
    v2f b;                                        // B fragment loaded ONCE
    b.x = W[(size_t)ka * DD + col];
    b.y = W[(size_t)(ka + 1) * DD + col];
#pragma unroll
    for (int s = 0; s < 4; ++s) {
      v2f a = *(const v2f*)(arow[s] + ka);
      acc[s] = wmma4(a, b, acc[s]);               // independent acc chains
    }
  }
  const float bv = bias[col];
#pragma unroll
  for (int s = 0; s < 4; ++s)
#pragma unroll
    for (int i = 0; i < 8; ++i)
      out[(size_t)(m0 + s * 16 + half * 8 + i) * DD + col] = acc[s][i] + bv;
}

// ---------------------------------------------------------------------------
// concepts[B*L, D]: fallback GEMM + where(found&valid, cls[widx], where(valid, fb, 0))
// One wave computes a 64x16 tile.  grid (B*L/64, D/16), block 32.
// ---------------------------------------------------------------------------
__global__ __launch_bounds__(32) void k_concepts(const int* __restrict__ ntid,
                                                 const int* __restrict__ nlen,
                                                 const int* __restrict__ t2i,
                                                 const float* __restrict__ emb,
                                                 const float* __restrict__ W,
                                                 const float* __restrict__ bias,
                                                 const float* __restrict__ cls,
                                                 float* __restrict__ concepts) {
  const int lane = threadIdx.x & 31;
  const int half = lane >> 4;
  const int lid  = lane & 15;
  const int m0 = blockIdx.x * 64, n0 = blockIdx.y * 16;
  const int col = n0 + lid;
  const float* arow[4];
#pragma unroll
  for (int s = 0; s < 4; ++s)
    arow[s] = emb + (size_t)ntid[m0 + s * 16 + lid] * WD;
  v8f acc[4];
#pragma unroll
  for (int s = 0; s < 4; ++s) acc[s] = (v8f){0.f,0.f,0.f,0.f,0.f,0.f,0.f,0.f};
  for (int k0 = 0; k0 < WD; k0 += 4) {
    const int ka = k0 + 2 * half;
    v2f b;
    b.x = W[(size_t)ka * DD + col];
    b.y = W[(size_t)(ka + 1) * DD + col];
#pragma unroll
    for (int s = 0; s < 4; ++s) {
      v2f a = *(const v2f*)(arow[s] + ka);
      acc[s] = wmma4(a, b, acc[s]);
    }
  }
  const float bv = bias[col];
#pragma unroll
  for (int s = 0; s < 4; ++s) {
#pragma unroll
    for (int i = 0; i < 8; ++i) {
      const int row  = m0 + s * 16 + half * 8 + i;   // 0..B*L-1
      const int bidx = row >> 4;                     // L == 16
      const int l    = row & 15;
      const int widx = t2i[ntid[row]];
      const bool valid = l < nlen[bidx];
      const bool found = widx > 0;
      float val = 0.f;
      if (valid) val = found ? cls[(size_t)widx * DD + col] : (acc[s][i] + bv);
      concepts[(size_t)row * DD + col] = val;
    }
  }
}

// ---------------------------------------------------------------------------
// out[B,D] = X[B,D] @ W[D,D] + bias, optional ReLU.
// One wave computes a 32x16 tile (2 M-subtiles).  grid (B/32, D/16), block 32.
// ---------------------------------------------------------------------------
__global__ __launch_bounds__(32) void k_gemm_xw(const float* __restrict__ X,
                                                const float* __restrict__ W,
                                                const float* __restrict__ bias,
                                                float* __restrict__ out, int relu) {
  const int lane = threadIdx.x & 31;
  const int half = lane >> 4;
  const int lid  = lane & 15;
  const int m0 = blockIdx.x * 32, n0 = blockIdx.y * 16;
  const int col = n0 + lid;
  const float* arow0 = X + (size_t)(m0 + lid) * DD;
  const float* arow1 = X + (size_t)(m0 + 16 + lid) * DD;
  v8f acc0 = {0.f,0.f,0.f,0.f,0.f,0.f,0.f,0.f};
  v8f acc1 = acc0;
  for (int k0 = 0; k0 < DD; k0 += 4) {
    const int ka = k0 + 2 * half;
    v2f b;
    b.x = W[(size_t)ka * DD + col];
    b.y = W[(size_t)(ka + 1) * DD + col];
    v2f a0 = *(const v2f*)(arow0 + ka);
    v2f a1 = *(const v2f*)(arow1 + ka);
    acc0 = wmma4(a0, b, acc0);
    acc1 = wmma4(a1, b, acc1);
  }
  const float bv = bias[col];
#pragma unroll
  for (int i = 0; i < 8; ++i) {
    float v0 = acc0[i] + bv;
    float v1 = acc1[i] + bv;
    if (relu) { v0 = fmaxf(v0, 0.f); v1 = fmaxf(v1, 0.f); }
    out[(size_t)(m0 + half * 8 + i) * DD + col] = v0;
    out[(size_t)(m0 + 16 + half * 8 + i) * DD + col] = v1;
  }
}

// ---------------------------------------------------------------------------
// In-place row softmax over D.  grid B, block 256.
// ---------------------------------------------------------------------------
__global__ void k_softmax(float* __restrict__ x) {
  __shared__ float red[256];
  float* p = x + (size_t)blockIdx.x * DD;
  const int tid = threadIdx.x;
  float mx = -3.402823466e+38f;
  for (int i = tid; i < DD; i += 256) mx = fmaxf(mx, p[i]);
  red[tid] = mx; __syncthreads();
  for (int s = 128; s > 0; s >>= 1) {
    if (tid < s) red[tid] = fmaxf(red[tid], red[tid + s]);
    __syncthreads();
  }
  mx = red[0]; __syncthreads();
  float sum = 0.f;
  for (int i = tid; i < DD; i += 256) { float e = __expf(p[i] - mx); p[i] = e; sum += e; }
  red[tid] = sum; __syncthreads();
  for (int s = 128; s > 0; s >>= 1) {
    if (tid < s) red[tid] += red[tid + s];
    __syncthreads();
  }
  const float inv = 1.f / red[0];
  for (int i = tid; i < DD; i += 256) p[i] *= inv;
}

// ---------------------------------------------------------------------------
// dn = row / ||row||;  optionally dsq = row*row.  grid B, block 256.
// ---------------------------------------------------------------------------
__global__ void k_row_l2(const float* __restrict__ src, float* __restrict__ dn,
                         float* __restrict__ dsq) {
  __shared__ float red[256];
  const int row = blockIdx.x, tid = threadIdx.x;
  const float* p = src + (size_t)row * DD;
  float s = 0.f;
  for (int i = tid; i < DD; i += 256) {
    const float v = p[i];
    s += v * v;
    if (dsq) dsq[(size_t)row * DD + i] = v * v;
  }
  red[tid] = s; __syncthreads();
  for (int st = 128; st > 0; st >>= 1) {
    if (tid < st) red[tid] += red[tid + st];
    __syncthreads();
  }
  const float inv = rsqrtf(red[0]);
  for (int i = tid; i < DD; i += 256) dn[(size_t)row * DD + i] = p[i] * inv;
}

// ---------------------------------------------------------------------------
// P = l2norm(trm_raw) * A_EM ;  Q = A_EM^2.  grid B, block 256.
// ---------------------------------------------------------------------------
__global__ void k_pq(const float* __restrict__ trm, const float* __restrict__ aem,
                     float* __restrict__ P, float* __restrict__ Q) {
  __shared__ float red[256];
  const int row = blockIdx.x, tid = threadIdx.x;
  const float* tp = trm + (size_t)row * DD;
  const float* ap = aem + (size_t)row * DD;
  float s = 0.f;
  for (int i = tid; i < DD; i += 256) { const float v = tp[i]; s += v * v; }
  red[tid] = s; __syncthreads();
  for (int st = 128; st > 0; st >>= 1) {
    if (tid < st) red[tid] += red[tid + st];
    __syncthreads();
  }
  const float inv = rsqrtf(red[0]);
  for (int i = tid; i < DD; i += 256) {
    const float a = ap[i];
    P[(size_t)row * DD + i] = tp[i] * inv * a;
    Q[(size_t)row * DD + i] = a * a;
  }
}

// ---------------------------------------------------------------------------
// score[b,c] = (P@t^T)/sqrt(Q@t2^T) + rn@tn^T.   grid (B/16, B/16), 3 WMMAs/step.
// ---------------------------------------------------------------------------
__global__ __launch_bounds__(32) void k_score(const float* __restrict__ P,
                                              const float* __restrict__ Q,
                                              const float* __restrict__ rn,
                                              const float* __restrict__ t,
                                              const float* __restrict__ t2,
                                              const float* __restrict__ tn,
                                              float* __restrict__ score) {
  const int lane = threadIdx.x & 31;
  const int half = lane >> 4;
  const int lid  = lane & 15;
  const int m0 = blockIdx.x * 16, n0 = blockIdx.y * 16;
  const int col = n0 + lid;
  const float* prow  = P  + (size_t)(m0 + lid) * DD;
  const float* qrow  = Q  + (size_t)(m0 + lid) * DD;
  const float* rrow  = rn + (size_t)(m0 + lid) * DD;
  const float* tcol  = t  + (size_t)col * DD;   // B[k][n] = t[n][k] (transposed GEMM)
  const float* t2col = t2 + (size_t)col * DD;
  const float* tncol = tn + (size_t)col * DD;
  v8f num = {0.f,0.f,0.f,0.f,0.f,0.f,0.f,0.f};
  v8f den = num, isv = num;
  for (int k0 = 0; k0 < DD; k0 += 4) {
    const int ka = k0 + 2 * half;
    v2f a1 = *(const v2f*)(prow + ka);
    v2f b1 = *(const v2f*)(tcol + ka);
    v2f a2 = *(const v2f*)(qrow + ka);
    v2f b2 = *(const v2f*)(t2col + ka);
    v2f a3 = *(const v2f*)(rrow + ka);
    v2f b3 = *(const v2f*)(tncol + ka);
    num = wmma4(a1, b1, num);   // three independent accumulator chains
    den = wmma4(a2, b2, den);
    isv = wmma4(a3, b3, isv);
  }
#pragma unroll
  for (int i = 0; i < 8; ++i) {
    const int row = m0 + half * 8 + i;
    score[(size_t)row * BB + col] = num[i] * rsqrtf(den[i]) + isv[i];
  }
}

extern "C" void kernel_launch(void* const* d_in, const int* in_sizes, int n_in,
                              void* d_out, int out_size, void* d_ws, size_t ws_size,
                              hipStream_t stream) {
  const float* r    = (const float*)d_in[0];
  const float* m    = (const float*)d_in[1];
  const float* t    = (const float*)d_in[2];
  const int*   ntid = (const int*)d_in[3];
  const int*   nlen = (const int*)d_in[4];
  const int*   antk = (const int*)d_in[5];
  const float* emb  = (const float*)d_in[6];
  const float* nW   = (const float*)d_in[7];
  const float* nb   = (const float*)d_in[8];
  const float* W1   = (const float*)d_in[9];
  const float* b1   = (const float*)d_in[10];
  const float* W2   = (const float*)d_in[11];
  const float* b2   = (const float*)d_in[12];
  const float* trW  = (const float*)d_in[13];
  const float* trb  = (const float*)d_in[14];
  const int*   t2i  = (const int*)d_in[15];

  float* score    = (float*)d_out;                 // [B,B]
  float* concepts = score + (size_t)BB * BB;       // [B,L,D]

  float* ws  = (float*)d_ws;
  float* cls = ws;  ws += (size_t)NN * DD;         // [N,D]
  float* h   = ws;  ws += (size_t)BB * DD;
  float* aem = ws;  ws += (size_t)BB * DD;
  float* trm = ws;  ws += (size_t)BB * DD;
  float* Pw  = ws;  ws += (size_t)BB * DD;
  float* Qw  = ws;  ws += (size_t)BB * DD;
  float* rnw = ws;  ws += (size_t)BB * DD;
  float* tnw = ws;  ws += (size_t)BB * DD;
  float* t2w = ws;  ws += (size_t)BB * DD;

  const dim3 wave(32);
  k_cls<<<dim3(NN / 64, DD / 16), wave, 0, stream>>>(antk, emb, nW, nb, cls);
  k_concepts<<<dim3(BB * LL / 64, DD / 16), wave, 0, stream>>>(ntid, nlen, t2i, emb,
                                                               nW, nb, cls, concepts);
  k_gemm_xw<<<dim3(BB / 32, DD / 16), wave, 0, stream>>>(m, W1, b1, h, 1);
  k_gemm_xw<<<dim3(BB / 32, DD / 16), wave, 0, stream>>>(h, W2, b2, aem, 0);
  k_softmax<<<BB, 256, 0, stream>>>(aem);
  k_gemm_xw<<<dim3(BB / 32, DD / 16), wave, 0, stream>>>(m, trW, trb, trm, 0);
  k_pq<<<BB, 256, 0, stream>>>(trm, aem, Pw, Qw);
  k_row_l2<<<BB, 256, 0, stream>>>(r, rnw, nullptr);
  k_row_l2<<<BB, 256, 0, stream>>>(t, tnw, t2w);
  k_score<<<dim3(BB / 16, BB / 16), wave, 0, stream>>>(Pw, Qw, rnw, t, t2w, tnw, score);
}